// _EquiAttention_257698038060
// MI455X (gfx1250) — compile-verified
//
#include <hip/hip_runtime.h>
#include <hip/hip_bf16.h>
#include <math.h>

typedef float v8f __attribute__((ext_vector_type(8)));
typedef float v2f __attribute__((ext_vector_type(2)));

#define NEG_BIAS (-1.0e10f)

// ---------------------------------------------------------------------------
// C[M,64] = X[M,64] * W[64,64]^T   (W row-major [out,in], torch Linear weight)
// One wave32 computes one 16x16 tile of C via V_WMMA_F32_16X16X4_F32,
// iterating K=64 in 16 chunks of 4.
//
// Fragment layouts per CDNA5 ISA 7.12.2:
//   A (16x4 f32, 2 VGPRs): lanes 0-15 -> M=lane, K={0,1}; lanes 16-31 -> K={2,3}
//   B (4x16 f32, 2 VGPRs): mirrored K split, N = lane%16
//   C (16x16 f32, 8 VGPRs): VGPR r -> M=r (lanes 0-15) / M=r+8 (lanes 16-31)
// ---------------------------------------------------------------------------
__global__ __launch_bounds__(256) void gemm_nt_wmma_f32(
    const float* __restrict__ X, const float* __restrict__ W,
    float* __restrict__ C, int M)
{
  const int lane      = threadIdx.x & 31;
  const int waveInBlk = threadIdx.x >> 5;
  const int wave      = blockIdx.x * 8 + waveInBlk;
  const int mt        = wave >> 2;   // 16-row tile of C
  const int nt        = wave & 3;    // 16-col tile of C (N=64 -> 4 tiles)
  if (mt * 16 >= M) return;          // wave-uniform; EXEC stays all-ones for WMMA

  const int mrow  = mt * 16 + (lane & 15);
  const int wrow  = nt * 16 + (lane & 15);
  const int khalf = (lane >> 4) << 1;          // 0 for lanes 0-15, 2 for 16-31

  const float* __restrict__ xp = X + (size_t)mrow * 64 + khalf;
  const float* __restrict__ wp = W + (size_t)wrow * 64 + khalf;

  v8f c = {};
#pragma unroll
  for (int kk = 0; kk < 64; kk += 4) {
    v2f a, b;
    a.x = xp[kk];  a.y = xp[kk + 1];
    b.x = wp[kk];  b.y = wp[kk + 1];
    // (neg_a, A, neg_b, B, c_mod, C, reuse_a, reuse_b)
    c = __builtin_amdgcn_wmma_f32_16x16x4_f32(false, a, false, b,
                                              (short)0, c, false, false);
  }

  const int crowBase = mt * 16 + ((lane >> 4) << 3);  // +8 for upper half-wave
  const int ccol     = nt * 16 + (lane & 15);
#pragma unroll
  for (int r = 0; r < 8; ++r)
    C[(size_t)(crowBase + r) * 64 + ccol] = c[r];
}

// ---------------------------------------------------------------------------
// One block per (b,q) row. Single streaming pass over messages[b,q,:,:]
// (the 268MB tensor -> read exactly once from HBM) with online softmax per
// channel dd, fused with the bkcd reduction against proj_equi (L2-resident)
// and the sum-of-squares needed for the L2-norm 'weights'.
// Thread map: dd = tid & 63 (channel), part = tid >> 6 (k-partition of 128).
// z[b,q,c,dd] = (acc_c / l) * sqrt(ss) / l
// ---------------------------------------------------------------------------
__global__ __launch_bounds__(256) void attn_softmax_kernel(
    const float* __restrict__ messages, const int* __restrict__ adj,
    const float* __restrict__ proj, float* __restrict__ z)
{
  __shared__ float s_bias[512];
  __shared__ float s_m[4][64], s_l[4][64], s_ss[4][64];
  __shared__ float s_a0[4][64], s_a1[4][64], s_a2[4][64];

  const int bq   = blockIdx.x;      // b*512 + q
  const int b    = bq >> 9;
  const int tid  = threadIdx.x;
  const int dd   = tid & 63;
  const int part = tid >> 6;

  // Stage the additive attention mask for this (b,q) row in LDS.
  for (int i = tid; i < 512; i += 256)
    s_bias[i] = (adj[(size_t)bq * 512 + i] > 0) ? 0.0f : NEG_BIAS;
  __syncthreads();

  const float* __restrict__ msg = messages + (size_t)bq * 512 * 64;
  const float* __restrict__ pb  = proj + (size_t)b * 512 * 3 * 64;

  float m = -1.0e30f, l = 0.0f, ss = 0.0f;
  float a0 = 0.0f, a1 = 0.0f, a2 = 0.0f;

  const int k0 = part * 128;
  for (int k = k0; k < k0 + 128; ++k) {
    // prefetch the message stream a few iterations ahead (global_prefetch_b8)
    __builtin_prefetch(&msg[(size_t)(k + 16) * 64 + dd], 0, 0);

    float x    = msg[(size_t)k * 64 + dd] + s_bias[k];
    float mn   = fmaxf(m, x);
    float corr = __expf(m - mn);     // 1.0 when max unchanged
    float e    = __expf(x - mn);
    float c2   = corr * corr;

    const float* __restrict__ pk = pb + (size_t)k * 3 * 64 + dd;
    float p0 = pk[0], p1 = pk[64], p2 = pk[128];

    l  = l  * corr + e;
    ss = ss * c2   + e * e;
    a0 = a0 * corr + e * p0;
    a1 = a1 * corr + e * p1;
    a2 = a2 * corr + e * p2;
    m  = mn;
  }

  s_m[part][dd]  = m;  s_l[part][dd]  = l;  s_ss[part][dd] = ss;
  s_a0[part][dd] = a0; s_a1[part][dd] = a1; s_a2[part][dd] = a2;
  __syncthreads();

  // Merge the 4 per-partition online-softmax states, then finalize.
  if (tid < 64) {
    float M0 = s_m[0][tid], L = s_l[0][tid], SS = s_ss[0][tid];
    float A0 = s_a0[0][tid], A1 = s_a1[0][tid], A2 = s_a2[0][tid];
#pragma unroll
    for (int p = 1; p < 4; ++p) {
      float m2 = s_m[p][tid];
      float mn = fmaxf(M0, m2);
      float f1 = __expf(M0 - mn);
      float f2 = __expf(m2 - mn);
      L  = L  * f1      + s_l[p][tid]  * f2;
      SS = SS * f1 * f1 + s_ss[p][tid] * f2 * f2;
      A0 = A0 * f1 + s_a0[p][tid] * f2;
      A1 = A1 * f1 + s_a1[p][tid] * f2;
      A2 = A2 * f1 + s_a2[p][tid] * f2;
      M0 = mn;
    }
    L = fmaxf(L, 1.0e-30f);              // guard fully-masked rows
    float scale = sqrtf(SS) / (L * L);   // (acc/L) * (sqrt(SS)/L)
    float* zo = z + (size_t)bq * 3 * 64 + tid;
    zo[0]   = A0 * scale;
    zo[64]  = A1 * scale;
    zo[128] = A2 * scale;
  }
}

// ---------------------------------------------------------------------------
extern "C" void kernel_launch(void* const* d_in, const int* in_sizes, int n_in,
                              void* d_out, int out_size, void* d_ws, size_t ws_size,
                              hipStream_t stream) {
  const float* v_equi   = (const float*)d_in[0];   // [B,Nkv,3,d]
  const float* messages = (const float*)d_in[1];   // [B,Nq,Nkv,d]
  const int*   adj      = (const int*)d_in[2];     // [B,Nq,Nkv]
  const float* w_coord  = (const float*)d_in[3];   // [d,d]
  const float* w_attn   = (const float*)d_in[4];   // [d,d]
  float* out = (float*)d_out;                      // [B,Nq,3,d]

  const int B = 4, Nq = 512, D = 64;
  const int Mproj = B * 512 * 3;   // 6144 rows: (b,k,c)
  const int Mout  = B * Nq * 3;    // 6144 rows: (b,q,c)

  float* proj = (float*)d_ws;                       // [B,Nkv,3,d] = 1.5 MB
  float* zbuf = proj + (size_t)Mproj * D;           // [B,Nq,3,d]  = 1.5 MB

  // waves = (M/16 m-tiles) * 4 n-tiles; 8 waves per 256-thread block
  const int gemmBlocks = ((Mproj / 16) * 4 + 7) / 8;  // 192

  // 1) proj_equi = v_equi @ w_coord^T  (WMMA f32)
  gemm_nt_wmma_f32<<<gemmBlocks, 256, 0, stream>>>(v_equi, w_coord, proj, Mproj);

  // 2) fused masked-softmax + bkcd reduction + row-norm -> z  (single HBM pass)
  attn_softmax_kernel<<<B * Nq, 256, 0, stream>>>(messages, adj, proj, zbuf);

  // 3) out = z @ w_attn^T  (WMMA f32)
  gemm_nt_wmma_f32<<<gemmBlocks, 256, 0, stream>>>(zbuf, w_attn, out, Mout);
}